// VoxelNeXtHead_22866405884092
// MI455X (gfx1250) — compile-verified
//
#include <hip/hip_runtime.h>

typedef __attribute__((ext_vector_type(2))) float v2f;
typedef __attribute__((ext_vector_type(8))) float v8f;

#define C_DIM 128
#define KTAPS 9
#define NHEAD 5
#define OUT_CH 11
#define BN_EPS 1e-5f
#define MTILE 32               // voxel rows per wave (2 x 16-row WMMA tiles)

struct HeadPtrs {
    const float* w[NHEAD];
    const float* b[NHEAD];
};

// One wave computes D[32 x 128] = gathered[32 x 1152] * W1h[1152 x 128]
// with fp32 WMMA 16x16x4. Two M-tiles share every B fragment (W1 reuse x2).
//
// fp32 A 16x4 layout: lane%16 = M row, lane-half selects K pair:
//   lanes 0-15:  VGPR0 -> K=0, VGPR1 -> K=1
//   lanes 16-31: VGPR0 -> K=2, VGPR1 -> K=3
// B 4x16 mirrors this: lane%16 = N column, lane-half selects K pair.
__device__ __forceinline__ void conv_tile32(const float* __restrict__ feats,
                                            const int* __restrict__ nbr,
                                            const float* __restrict__ W1h,
                                            int row0, int Nv, int lane,
                                            v8f accA[8], v8f accB[8]) {
    const int m    = lane & 15;
    const int half = lane >> 4;           // 0 or 1
    int rowA = row0 + m;        if (rowA >= Nv) rowA = Nv - 1;
    int rowB = row0 + 16 + m;   if (rowB >= Nv) rowB = Nv - 1;

    for (int tap = 0; tap < KTAPS; ++tap) {
        const int rA = nbr[rowA * KTAPS + tap];
        const int rB = nbr[rowB * KTAPS + tap];
        const float* __restrict__ arowA = feats + (size_t)rA * C_DIM;
        const float* __restrict__ arowB = feats + (size_t)rB * C_DIM;
        const float* __restrict__ wtap  = W1h + (size_t)tap * C_DIM * C_DIM;

        #pragma unroll 2
        for (int c0 = 0; c0 < C_DIM; c0 += 4) {
            const int k = c0 + 2 * half;  // this lane-half's K base

            // ---- load phase: A (2 x b64) + all 8 B fragments (16 x b32) ----
            v2f a0, a1, b[8];
            a0.x = arowA[k];
            a0.y = arowA[k + 1];
            a1.x = arowB[k];
            a1.y = arowB[k + 1];

            const float* __restrict__ wk = wtap + (size_t)k * C_DIM + m;
            #pragma unroll
            for (int nt = 0; nt < 8; ++nt) {
                b[nt].x = wk[nt * 16];            // W1[k  ][nt*16 + m]
                b[nt].y = wk[nt * 16 + C_DIM];    // W1[k+1][nt*16 + m]
            }

            // ---- compute phase: 16 WMMAs on the prefetched fragments ----
            #pragma unroll
            for (int nt = 0; nt < 8; ++nt) {
                accA[nt] = __builtin_amdgcn_wmma_f32_16x16x4_f32(
                    false, a0, false, b[nt], (short)0, accA[nt], false, false);
                accB[nt] = __builtin_amdgcn_wmma_f32_16x16x4_f32(
                    false, a1, false, b[nt], (short)0, accB[nt], false, false);
            }
        }
    }
}

// Pass 1: GEMM tile + per-(head, channel) sum / sum-of-squares accumulation.
__global__ void __launch_bounds__(32, 4)
k_conv_stats(const float* __restrict__ feats,
             const int* __restrict__ nbr,
             const float* __restrict__ W1,
             float* __restrict__ sum,
             float* __restrict__ sumsq,
             int Nv) {
    const int tile = blockIdx.x;
    const int h    = blockIdx.y;
    const int lane = threadIdx.x;
    const int row0 = tile * MTILE;

    v8f accA[8] = {};
    v8f accB[8] = {};
    conv_tile32(feats, nbr, W1 + (size_t)h * KTAPS * C_DIM * C_DIM,
                row0, Nv, lane, accA, accB);

    // C/D layout: VGPR j = rows {j, j+8}; lane%16 = column; lane-half = row half.
    const int nb = lane & 15;
    #pragma unroll
    for (int nt = 0; nt < 8; ++nt) {
        float s = 0.f, q = 0.f;
        #pragma unroll
        for (int j = 0; j < 8; ++j) {
            const float va = accA[nt][j];
            const float vb = accB[nt][j];
            s += va + vb;
            q += va * va + vb * vb;
        }
        const int d = nt * 16 + nb;
        atomicAdd(&sum  [h * C_DIM + d], s);
        atomicAdd(&sumsq[h * C_DIM + d], q);
    }
}

// Pass 2: recompute GEMM tile, BatchNorm(+ReLU) with pass-1 stats, then the
// per-head [128 -> oc] projection from an LDS-staged tile, write [N,11] slice.
__global__ void __launch_bounds__(32, 4)
k_bn_head(const float* __restrict__ feats,
          const int* __restrict__ nbr,
          const float* __restrict__ W1,
          const float* __restrict__ gamma,
          const float* __restrict__ beta,
          const float* __restrict__ sum,
          const float* __restrict__ sumsq,
          HeadPtrs hp,
          float* __restrict__ out,
          int Nv) {
    __shared__ float yt[MTILE * 132];     // padded stride: bank-conflict free

    const int tile = blockIdx.x;
    const int h    = blockIdx.y;
    const int lane = threadIdx.x;
    const int row0 = tile * MTILE;

    v8f accA[8] = {};
    v8f accB[8] = {};
    conv_tile32(feats, nbr, W1 + (size_t)h * KTAPS * C_DIM * C_DIM,
                row0, Nv, lane, accA, accB);

    const float inv_n = 1.0f / (float)Nv;
    const int nb   = lane & 15;
    const int half = lane >> 4;

    #pragma unroll
    for (int nt = 0; nt < 8; ++nt) {
        const int d = nt * 16 + nb;
        const float mu = sum[h * C_DIM + d] * inv_n;
        float var = sumsq[h * C_DIM + d] * inv_n - mu * mu;
        var = fmaxf(var, 0.f);
        const float istd = rsqrtf(var + BN_EPS);
        const float g  = gamma[h * C_DIM + d];
        const float bb = beta [h * C_DIM + d];
        #pragma unroll
        for (int j = 0; j < 8; ++j) {
            const int mr = j + 8 * half;
            const float va = (accA[nt][j] - mu) * istd * g + bb;
            const float vb = (accB[nt][j] - mu) * istd * g + bb;
            yt[ mr       * 132 + d] = fmaxf(va, 0.f);
            yt[(mr + 16) * 132 + d] = fmaxf(vb, 0.f);
        }
    }
    __syncthreads();

    const int OC [NHEAD] = {3, 2, 1, 3, 2};
    const int OFF[NHEAD] = {0, 3, 5, 6, 9};
    const int oc  = OC[h];
    const int off = OFF[h];
    const float* __restrict__ wh = hp.w[h];
    const float* __restrict__ bh = hp.b[h];

    const int m = lane & 15;
    for (int mt = 0; mt < 2; ++mt) {
        const int r = mt * 16 + m;
        if (row0 + r < Nv) {
            for (int o0 = 0; o0 < oc; o0 += 2) {
                const int o = o0 + half;  // lane-halves split output channels
                if (o < oc) {
                    float a = 0.f;
                    #pragma unroll 8
                    for (int d = 0; d < C_DIM; ++d)
                        a += yt[r * 132 + d] * wh[d * oc + o];
                    out[(size_t)(row0 + r) * OUT_CH + off + o] = a + bh[o];
                }
            }
        }
    }
}

__global__ void k_zero(float* __restrict__ p, int n) {
    const int i = blockIdx.x * blockDim.x + threadIdx.x;
    if (i < n) p[i] = 0.f;
}

extern "C" void kernel_launch(void* const* d_in, const int* in_sizes, int n_in,
                              void* d_out, int out_size, void* d_ws, size_t ws_size,
                              hipStream_t stream) {
    const float* feats = (const float*)d_in[0];
    const int*   nbr   = (const int*)  d_in[1];
    const float* W1    = (const float*)d_in[2];
    const float* gamma = (const float*)d_in[3];
    const float* beta  = (const float*)d_in[4];

    HeadPtrs hp;
    for (int i = 0; i < NHEAD; ++i) {
        hp.w[i] = (const float*)d_in[5 + 2 * i];
        hp.b[i] = (const float*)d_in[6 + 2 * i];
    }

    const int Nv    = in_sizes[0] / C_DIM;          // 40000
    const int tiles = (Nv + MTILE - 1) / MTILE;     // 1250

    float* sum   = (float*)d_ws;                    // [5][128]
    float* sumsq = sum + NHEAD * C_DIM;             // [5][128]

    k_zero<<<dim3((2 * NHEAD * C_DIM + 255) / 256), 256, 0, stream>>>(
        sum, 2 * NHEAD * C_DIM);

    dim3 grid(tiles, NHEAD);
    k_conv_stats<<<grid, 32, 0, stream>>>(feats, nbr, W1, sum, sumsq, Nv);
    k_bn_head  <<<grid, 32, 0, stream>>>(feats, nbr, W1, gamma, beta,
                                         sum, sumsq, hp, (float*)d_out, Nv);
}